// CausalNCMomentumAttention_v3_37271726195007
// MI455X (gfx1250) — compile-verified
//
#include <hip/hip_runtime.h>
#include <hip/hip_bf16.h>

// Problem constants (from reference setup_inputs)
#define NB 2        // batch N
#define LL 2048     // sequence length
#define HH 8        // heads
#define EE 64       // key dim
#define MM 64       // value dim
#define CK 64       // chunk length
#define NC (LL/CK)  // chunks per sequence = 32
#define EPS 1e-6f
#define PAD 4
#define LDW (EE + PAD)   // 68 floats/row -> 272B rows: 16B aligned + bank-shifted

typedef __attribute__((ext_vector_type(2))) float v2f;
typedef __attribute__((ext_vector_type(8))) float v8f;

__device__ __forceinline__ float fmap(float x) {
    // elu(x)+1 : x>0 ? x+1 : exp(x)
    return x > 0.0f ? x + 1.0f : __expf(x);
}

// ---------------------------------------------------------------------------
// Kernel 1: momentum coefficient mu[n][l], l in [0, L-2]
//   ||K_l (x) V_l||^2   = sum_h ||K||^2 ||V||^2
//   <K1(x)V1, K0(x)V0>  = sum_h (K1.K0)(V1.V0)
// ---------------------------------------------------------------------------
__global__ void mu_kernel(const float* __restrict__ keys,
                          const float* __restrict__ values,
                          const float* __restrict__ klen,
                          float* __restrict__ mu) {
    int b = blockIdx.x;             // over N*(L-1)
    int n = b / (LL - 1);
    int l = b % (LL - 1);
    int t = threadIdx.x;            // 256 threads: wave h handles head h
    int h = t >> 5;
    int lane = t & 31;

    float kl0 = klen[n * LL + l];
    float kl1 = klen[n * LL + l + 1];

    float kn0 = 0.f, kn1 = 0.f, kd = 0.f, vn0 = 0.f, vn1 = 0.f, vd = 0.f;
    for (int e = lane; e < EE; e += 32) {
        float k0 = fmap(keys[(((size_t)n * LL + l)     * HH + h) * EE + e]) * kl0;
        float k1 = fmap(keys[(((size_t)n * LL + l + 1) * HH + h) * EE + e]) * kl1;
        kn0 += k0 * k0; kn1 += k1 * k1; kd += k0 * k1;
        float v0 = values[(((size_t)n * LL + l)     * HH + h) * MM + e];
        float v1 = values[(((size_t)n * LL + l + 1) * HH + h) * MM + e];
        vn0 += v0 * v0; vn1 += v1 * v1; vd += v0 * v1;
    }
    for (int off = 16; off > 0; off >>= 1) {
        kn0 += __shfl_xor(kn0, off, 32);
        kn1 += __shfl_xor(kn1, off, 32);
        kd  += __shfl_xor(kd,  off, 32);
        vn0 += __shfl_xor(vn0, off, 32);
        vn1 += __shfl_xor(vn1, off, 32);
        vd  += __shfl_xor(vd,  off, 32);
    }
    __shared__ float sh[HH][6];
    if (lane == 0) {
        sh[h][0] = kn0; sh[h][1] = kn1; sh[h][2] = kd;
        sh[h][3] = vn0; sh[h][4] = vn1; sh[h][5] = vd;
    }
    __syncthreads();
    if (t == 0) {
        float base2 = 0.f, diff2 = 0.f;
        for (int hh = 0; hh < HH; ++hh) {
            float a = sh[hh][0], b1 = sh[hh][1], c = sh[hh][2];
            float d = sh[hh][3], e1 = sh[hh][4], f = sh[hh][5];
            base2 += a * d;
            diff2 += b1 * e1 + a * d - 2.0f * c * f;
        }
        diff2 = fmaxf(diff2, 0.0f);
        float ratio = sqrtf(diff2) / sqrtf(base2);   // ||diff|| / ||base||
        float r4 = sqrtf(ratio);
        float m_ = 1.0f - r4;
        m_ = m_ * m_;
        m_ = fminf(fmaxf(m_, 0.0f), 0.9999f);
        mu[n * LL + l] = m_;
    }
}

// ---------------------------------------------------------------------------
// Kernel 2: reverse affine scan   w[L-1]=1; w[i]=w[i+1]*mu[i]+1
// ---------------------------------------------------------------------------
__global__ void weights_kernel(const float* __restrict__ mu,
                               float* __restrict__ weights) {
    int n = blockIdx.x;
    if (threadIdx.x != 0) return;
    float w = 1.0f;
    weights[n * LL + LL - 1] = 1.0f;
    for (int i = LL - 2; i >= 0; --i) {
        w = w * mu[n * LL + i] + 1.0f;
        weights[n * LL + i] = w;
    }
}

// ---------------------------------------------------------------------------
// Kernel 3: per-chunk state deltas  dS[E][M] = K^T @ (diag(w) V)   (WMMA f32)
//           plus per-chunk raw-K column sums (for Z normalizer)
// ---------------------------------------------------------------------------
__global__ void sdelta_kernel(const float* __restrict__ keys,
                              const float* __restrict__ values,
                              const float* __restrict__ klen,
                              const float* __restrict__ weights,
                              float* __restrict__ Sbuf,
                              float* __restrict__ Ksum) {
    __shared__ float Klds[CK][LDW];   // raw featuremapped K
    __shared__ float Vlds[CK][LDW];   // V pre-scaled by w[s]

    int b = blockIdx.x;               // over N*H*NC
    int c = b % NC;
    int h = (b / NC) % HH;
    int n = b / (NC * HH);
    int c0 = c * CK;
    int t = threadIdx.x;              // 256

    for (int i = t; i < CK * EE / 4; i += 256) {
        int s  = i >> 4;
        int e4 = (i & 15) * 4;
        int l  = c0 + s;
        float kl = klen[n * LL + l];
        float w  = weights[n * LL + l];
        const float4 kv = *(const float4*)&keys  [(((size_t)n * LL + l) * HH + h) * EE + e4];
        const float4 vv = *(const float4*)&values[(((size_t)n * LL + l) * HH + h) * MM + e4];
        float4 ko, vo;
        ko.x = fmap(kv.x) * kl; ko.y = fmap(kv.y) * kl;
        ko.z = fmap(kv.z) * kl; ko.w = fmap(kv.w) * kl;
        vo.x = vv.x * w; vo.y = vv.y * w; vo.z = vv.z * w; vo.w = vv.w * w;
        *(float4*)&Klds[s][e4] = ko;
        *(float4*)&Vlds[s][e4] = vo;
    }
    __syncthreads();

    if (t < EE) {  // raw-K chunk column sums
        float acc = 0.f;
        for (int s = 0; s < CK; ++s) acc += Klds[s][t];
        Ksum[(((size_t)n * HH + h) * NC + c) * EE + t] = acc;
    }

    int wave = t >> 5, lane = t & 31, lr = lane & 15, lh = lane >> 4;
    float* out = Sbuf + (((size_t)n * HH + h) * NC + c) * (EE * MM);

    for (int tile = wave; tile < 16; tile += 8) {
        int ti = tile >> 2, tj = tile & 3;
        v8f acc = {};
        for (int k0 = 0; k0 < CK; k0 += 4) {
            int s = k0 + 2 * lh;
            v2f a, bb;
            int e = ti * 16 + lr;               // A = K^T : A[e][s]
            a.x = Klds[s][e];
            a.y = Klds[s + 1][e];
            int m = tj * 16 + lr;               // B = Vw  : B[s][m]
            bb.x = Vlds[s][m];
            bb.y = Vlds[s + 1][m];
            acc = __builtin_amdgcn_wmma_f32_16x16x4_f32(false, a, false, bb,
                                                        (short)0, acc, false, false);
        }
        for (int r = 0; r < 8; ++r) {
            int row = ti * 16 + r + 8 * lh;
            int col = tj * 16 + lr;
            out[row * MM + col] = acc[r];
        }
    }
}

// ---------------------------------------------------------------------------
// Kernel 4: exclusive prefix over chunks (states + Ksum), per (n,h)
// ---------------------------------------------------------------------------
__global__ void prefix_kernel(float* __restrict__ Sbuf,
                              float* __restrict__ Ksum) {
    int b = blockIdx.x;               // over N*H
    int t = threadIdx.x;
    float4* base = (float4*)(Sbuf + (size_t)b * NC * (EE * MM));
    for (int i = t; i < EE * MM / 4; i += 256) {
        float4 acc = {0.f, 0.f, 0.f, 0.f};
        for (int c = 0; c < NC; ++c) {
            float4 v = base[(size_t)c * (EE * MM / 4) + i];
            base[(size_t)c * (EE * MM / 4) + i] = acc;
            acc.x += v.x; acc.y += v.y; acc.z += v.z; acc.w += v.w;
        }
    }
    if (t < EE) {
        float* kb = Ksum + (size_t)b * NC * EE;
        float acc = 0.f;
        for (int c = 0; c < NC; ++c) {
            float v = kb[c * EE + t];
            kb[c * EE + t] = acc;
            acc += v;
        }
    }
}

// ---------------------------------------------------------------------------
// Kernel 5: output pass per (n,h,chunk):
//   P_raw = mask(Q @ K^T)                    (WMMA, masked into LDS)
//   zden  = rowsum(P_raw) + Q . KsumPrefix + eps
//   out   = (Q @ Sprefix + P_raw @ (diag(w) V)) / zden
// ---------------------------------------------------------------------------
__global__ void out_kernel(const float* __restrict__ queries,
                           const float* __restrict__ keys,
                           const float* __restrict__ values,
                           const float* __restrict__ klen,
                           const float* __restrict__ weights,
                           const float* __restrict__ Sbuf,
                           const float* __restrict__ Ksum,
                           float* __restrict__ out) {
    __shared__ float Qlds[CK][LDW];
    __shared__ float KP[CK][LDW];     // holds K first, then masked P
    __shared__ float Vlds[CK][LDW];   // V pre-scaled by w[s]
    __shared__ float Slds[EE][LDW];   // S prefix tile
    __shared__ float ks[EE];
    __shared__ float zden[CK];

    int b = blockIdx.x;               // over N*H*NC
    int c = b % NC;
    int h = (b / NC) % HH;
    int n = b / (NC * HH);
    int c0 = c * CK;
    int t = threadIdx.x;              // 256

    const float* Sp = Sbuf + (((size_t)n * HH + h) * NC + c) * (EE * MM);
    for (int i = t; i < CK * EE / 4; i += 256) {
        int s  = i >> 4;
        int e4 = (i & 15) * 4;
        int l  = c0 + s;
        float kl = klen[n * LL + l];
        float w  = weights[n * LL + l];
        const float4 qv = *(const float4*)&queries[(((size_t)n * LL + l) * HH + h) * EE + e4];
        const float4 kv = *(const float4*)&keys   [(((size_t)n * LL + l) * HH + h) * EE + e4];
        const float4 vv = *(const float4*)&values [(((size_t)n * LL + l) * HH + h) * MM + e4];
        float4 qo, ko, vo;
        qo.x = fmap(qv.x);      qo.y = fmap(qv.y);
        qo.z = fmap(qv.z);      qo.w = fmap(qv.w);
        ko.x = fmap(kv.x) * kl; ko.y = fmap(kv.y) * kl;
        ko.z = fmap(kv.z) * kl; ko.w = fmap(kv.w) * kl;
        vo.x = vv.x * w; vo.y = vv.y * w; vo.z = vv.z * w; vo.w = vv.w * w;
        *(float4*)&Qlds[s][e4] = qo;
        *(float4*)&KP[s][e4]   = ko;
        *(float4*)&Vlds[s][e4] = vo;
        *(float4*)&Slds[s][e4] = *(const float4*)&Sp[s * MM + e4];
    }
    if (t < EE) ks[t] = Ksum[(((size_t)n * HH + h) * NC + c) * EE + t];
    __syncthreads();

    int wave = t >> 5, lane = t & 31, lr = lane & 15, lh = lane >> 4;

    // ---- P = Q @ K^T, two 16x16 tiles per wave, kept in registers ----
    v8f p0 = {}, p1 = {};
    for (int half = 0; half < 2; ++half) {
        int tile = wave + half * 8;
        int ti = tile >> 2, tj = tile & 3;
        v8f acc = {};
        for (int k0 = 0; k0 < EE; k0 += 4) {
            int e = k0 + 2 * lh;
            v2f a, bb;
            int row = ti * 16 + lr;             // A = Q : A[l][e]
            a.x = Qlds[row][e];
            a.y = Qlds[row][e + 1];
            int col = tj * 16 + lr;             // B = K^T : B[e][s] = K[s][e]
            bb.x = KP[col][e];
            bb.y = KP[col][e + 1];
            acc = __builtin_amdgcn_wmma_f32_16x16x4_f32(false, a, false, bb,
                                                        (short)0, acc, false, false);
        }
        if (half) p1 = acc; else p0 = acc;
    }
    __syncthreads();   // everyone finished reading K from KP

    // ---- write causally-masked P into KP ----
    for (int half = 0; half < 2; ++half) {
        v8f acc = half ? p1 : p0;
        int tile = wave + half * 8;
        int ti = tile >> 2, tj = tile & 3;
        for (int r = 0; r < 8; ++r) {
            int row = ti * 16 + r + 8 * lh;
            int col = tj * 16 + lr;
            KP[row][col] = (col <= row) ? acc[r] : 0.0f;
        }
    }
    __syncthreads();

    // ---- Z denominator per row ----
    if (t < CK) {
        float s1 = 0.f;
        for (int s = 0; s <= t; ++s) s1 += KP[t][s];
        float s2 = 0.f;
        for (int e = 0; e < EE; ++e) s2 += Qlds[t][e] * ks[e];
        zden[t] = s1 + s2 + EPS;
    }
    __syncthreads();

    // ---- out = Q @ Sprefix + P @ Vw ----
    for (int half = 0; half < 2; ++half) {
        int tile = wave + half * 8;
        int ti = tile >> 2, tj = tile & 3;
        v8f acc = {};
        for (int k0 = 0; k0 < EE; k0 += 4) {   // GEMM1: Q @ Sprefix
            int e = k0 + 2 * lh;
            v2f a, bb;
            int row = ti * 16 + lr;
            a.x = Qlds[row][e];
            a.y = Qlds[row][e + 1];
            int col = tj * 16 + lr;
            bb.x = Slds[e][col];
            bb.y = Slds[e + 1][col];
            acc = __builtin_amdgcn_wmma_f32_16x16x4_f32(false, a, false, bb,
                                                        (short)0, acc, false, false);
        }
        for (int k0 = 0; k0 < CK; k0 += 4) {   // GEMM2: P @ Vw
            int s = k0 + 2 * lh;
            v2f a, bb;
            int row = ti * 16 + lr;
            a.x = KP[row][s];
            a.y = KP[row][s + 1];
            int col = tj * 16 + lr;
            bb.x = Vlds[s][col];
            bb.y = Vlds[s + 1][col];
            acc = __builtin_amdgcn_wmma_f32_16x16x4_f32(false, a, false, bb,
                                                        (short)0, acc, false, false);
        }
        for (int r = 0; r < 8; ++r) {
            int row = ti * 16 + r + 8 * lh;
            int col = tj * 16 + lr;
            int l = c0 + row;
            out[(((size_t)n * LL + l) * HH + h) * MM + col] = acc[r] / zden[row];
        }
    }
}

// ---------------------------------------------------------------------------
extern "C" void kernel_launch(void* const* d_in, const int* in_sizes, int n_in,
                              void* d_out, int out_size, void* d_ws, size_t ws_size,
                              hipStream_t stream) {
    const float* queries = (const float*)d_in[0];
    const float* keys    = (const float*)d_in[1];
    const float* values  = (const float*)d_in[2];
    const float* klen    = (const float*)d_in[3];
    float* outp = (float*)d_out;

    // workspace layout (floats)
    float* ws      = (float*)d_ws;
    float* Sbuf    = ws;                                    // N*H*NC*E*M = 2,097,152
    float* Ksum    = Sbuf + (size_t)NB * HH * NC * EE * MM; // N*H*NC*E  =    32,768
    float* mu      = Ksum + (size_t)NB * HH * NC * EE;      // N*L       =     4,096
    float* weights = mu + (size_t)NB * LL;                  // N*L       =     4,096

    mu_kernel<<<NB * (LL - 1), 256, 0, stream>>>(keys, values, klen, mu);
    weights_kernel<<<NB, 32, 0, stream>>>(mu, weights);
    sdelta_kernel<<<NB * HH * NC, 256, 0, stream>>>(keys, values, klen, weights, Sbuf, Ksum);
    prefix_kernel<<<NB * HH, 256, 0, stream>>>(Sbuf, Ksum);
    out_kernel<<<NB * HH * NC, 256, 0, stream>>>(queries, keys, values, klen,
                                                 weights, Sbuf, Ksum, outp);
}